// GCNDecoder_64974265254503
// MI455X (gfx1250) — compile-verified
//
#include <hip/hip_runtime.h>

// ---------------------------------------------------------------------------
// GCN decoder for MI455X (gfx1250, wave32).
// GEMMs use V_WMMA_F32_16X16X4_F32 (exact fp32, matches reference dtype).
// Aggregation (the roofline bottleneck, ~70 GB of random traffic) uses
// float4 gathers + global_atomic_add_f32 scatters.
// ---------------------------------------------------------------------------

typedef float v2f __attribute__((ext_vector_type(2)));
typedef float v8f __attribute__((ext_vector_type(8)));

// ---------------- graph normalization ----------------

__global__ void k_deg_init(float* __restrict__ deg, int n) {
  int i = blockIdx.x * blockDim.x + threadIdx.x;
  if (i < n) deg[i] = 1.0f;  // self-loop weight
}

__global__ void k_deg_accum(const int* __restrict__ dst, const float* __restrict__ ew,
                            float* __restrict__ deg, int e) {
  int i = blockIdx.x * blockDim.x + threadIdx.x;
  if (i < e) atomicAdd(&deg[dst[i]], ew[i]);
}

__global__ void k_make_dis(float* __restrict__ degdis, int n) {
  int i = blockIdx.x * blockDim.x + threadIdx.x;
  if (i < n) {
    float d = degdis[i];
    degdis[i] = (d > 0.0f) ? rsqrtf(d) : 0.0f;
  }
}

__global__ void k_make_norm(const int* __restrict__ src, const int* __restrict__ dst,
                            const float* __restrict__ ew, const float* __restrict__ dis,
                            float* __restrict__ nrm, int e) {
  int i = blockIdx.x * blockDim.x + threadIdx.x;
  if (i < e) nrm[i] = dis[src[i]] * ew[i] * dis[dst[i]];
}

// ---------------- fp32 WMMA GEMM: C[M,Nout] = A[M,K] * B[K,Nout] ----------------
// One wave per block. blockIdx.x -> 16-row M tile, blockIdx.y -> NT*16-col strip.
// NT is a compile-time constant so the NT accumulators are unconditionally
// updated in place. Per k-step: issue ALL fragment loads first (distinct
// locals -> one load clause / one wait), then NT back-to-back WMMAs; the
// scheduler can overlap the next unrolled k-step's loads with these WMMAs.
//
// Fragment layouts (CDNA5 ISA 7.12.2, 32-bit operands, wave32):
//   A 16x4 : lanes 0-15 -> M=lane, VGPR0=K0,VGPR1=K1; lanes 16-31 -> K2,K3
//   B 4x16 : VGPR0 -> K=2*half, VGPR1 -> K=2*half+1, N = lane&15
//   C 16x16: VGPR v -> row v + 8*half, col = lane&15
template <int NT>
__global__ __launch_bounds__(32) void k_gemm_wmma_f32(
    const float* __restrict__ A, const float* __restrict__ B, float* __restrict__ C,
    int K, int Nout, int nbase) {
  const int lane = threadIdx.x & 31;
  const int half = lane >> 4;
  const int mr   = lane & 15;
  const int m0 = blockIdx.x << 4;
  const int n0 = nbase + blockIdx.y * (NT * 16);

  v8f acc[NT];
#pragma unroll
  for (int t = 0; t < NT; ++t) acc[t] = v8f{};

  const float* Arow = A + (size_t)(m0 + mr) * K + 2 * half;
  const float* Bp   = B + (size_t)(2 * half) * Nout + n0 + mr;
  const size_t bstep = (size_t)4 * Nout;

#pragma unroll 4
  for (int k0 = 0; k0 < K; k0 += 4) {
    // ---- phase 1: all loads for this k-step (A b64 + 2*NT B dwords) ----
    v2f a = *(const v2f*)(Arow + k0);       // K = k0+2*half, k0+2*half+1 of row m
    v2f b[NT];
#pragma unroll
    for (int t = 0; t < NT; ++t) {
      b[t].x = Bp[t * 16];                  // row K = k0 + 2*half
      b[t].y = Bp[(size_t)Nout + t * 16];   // row K = k0 + 2*half + 1
    }
    Bp += bstep;
    // ---- phase 2: NT back-to-back WMMAs ----
#pragma unroll
    for (int t = 0; t < NT; ++t) {
      acc[t] = __builtin_amdgcn_wmma_f32_16x16x4_f32(
          false, a, false, b[t], (short)0, acc[t], false, false);
    }
  }

#pragma unroll
  for (int t = 0; t < NT; ++t) {
    float* Cp = C + (size_t)(m0 + half * 8) * Nout + n0 + t * 16 + mr;
#pragma unroll
    for (int v = 0; v < 8; ++v) Cp[(size_t)v * Nout] = acc[t][v];
  }
}

// ---------------- out = bias + dis^2 * H   (self-loop term, full overwrite) ----
__global__ void k_init_out(const float* __restrict__ H, const float* __restrict__ bias,
                           const float* __restrict__ dis, float* __restrict__ out,
                           int n, int fout) {
  const int cpe = fout >> 2;                 // float4 chunks per row
  int idx = blockIdx.x * blockDim.x + threadIdx.x;
  if (idx >= n * cpe) return;
  int i = idx / cpe;
  int c = idx - i * cpe;
  float s = dis[i];
  s *= s;                                    // = 1/deg  (self-loop norm)
  float4 hv = ((const float4*)H)[(size_t)i * cpe + c];
  float4 bv = ((const float4*)bias)[c];
  float4 o;
  o.x = bv.x + s * hv.x;
  o.y = bv.y + s * hv.y;
  o.z = bv.z + s * hv.z;
  o.w = bv.w + s * hv.w;
  ((float4*)out)[(size_t)i * cpe + c] = o;
}

// ---------------- edge scatter: out[dst] += norm_e * H[src] ----------------
// epb edges per 256-thread block; tpe = 256/epb threads stride the feature dim.
__global__ void k_edge_scatter(const float* __restrict__ H, float* __restrict__ out,
                               const int* __restrict__ src, const int* __restrict__ dst,
                               const float* __restrict__ nrm, int e, int fout, int epb) {
  const int cpe = fout >> 2;
  const int tpe = blockDim.x / epb;
  const int le  = threadIdx.x / tpe;
  const int c0  = threadIdx.x - le * tpe;
  const int edge = blockIdx.x * epb + le;
  if (edge >= e) return;
  const int s = src[edge];
  const int d = dst[edge];
  const float w = nrm[edge];
  const float4* h4 = (const float4*)H + (size_t)s * cpe;
  float* op = out + (size_t)d * fout;
  for (int c = c0; c < cpe; c += tpe) {
    float4 h = h4[c];
    atomicAdd(op + 4 * c + 0, w * h.x);
    atomicAdd(op + 4 * c + 1, w * h.y);
    atomicAdd(op + 4 * c + 2, w * h.z);
    atomicAdd(op + 4 * c + 3, w * h.w);
  }
}

// ---------------------------------------------------------------------------

extern "C" void kernel_launch(void* const* d_in, const int* in_sizes, int n_in,
                              void* d_out, int out_size, void* d_ws, size_t ws_size,
                              hipStream_t stream) {
  const float* x  = (const float*)d_in[0];
  const int*   src = (const int*)d_in[1];
  const int*   dst = (const int*)d_in[2];
  const float* ew = (const float*)d_in[3];
  const float* W1 = (const float*)d_in[4];
  const float* b1 = (const float*)d_in[5];
  const float* W2 = (const float*)d_in[6];
  const float* b2 = (const float*)d_in[7];
  const float* Wr = (const float*)d_in[8];
  const float* br = (const float*)d_in[9];
  float* out = (float*)d_out;

  const int D_IN  = 64;
  const int N     = in_sizes[0] / D_IN;
  const int E     = in_sizes[1];
  const int D_H1  = in_sizes[5];
  const int D_H2  = in_sizes[7];
  const int D_OUT = in_sizes[9];

  // workspace carve-up (~1.02 GB): dis | norm | H (pre-agg) | A (aggregated)
  auto al = [](size_t v) { return (v + 255) & ~(size_t)255; };
  char* ws = (char*)d_ws;
  size_t off = 0;
  float* dis  = (float*)(ws + off); off = al(off + (size_t)N * 4);
  float* nrm  = (float*)(ws + off); off = al(off + (size_t)E * 4);
  float* hbuf = (float*)(ws + off); off = al(off + (size_t)N * D_OUT * 4);
  float* abuf = (float*)(ws + off); off = al(off + (size_t)N * D_H2 * 4);
  (void)ws_size; (void)n_in; (void)out_size;

  const int T = 256;

  // degree / normalization (layer-invariant, computed once)
  k_deg_init <<<(N + T - 1) / T, T, 0, stream>>>(dis, N);
  k_deg_accum<<<(E + T - 1) / T, T, 0, stream>>>(dst, ew, dis, E);
  k_make_dis <<<(N + T - 1) / T, T, 0, stream>>>(dis, N);
  k_make_norm<<<(E + T - 1) / T, T, 0, stream>>>(src, dst, ew, dis, nrm, E);

  auto layer = [&](const float* Xin, const float* W, const float* b,
                   float* Aout, int K, int F) {
    const int mt   = N / 16;
    const int main = F / 64;                  // full 64-col strips
    const int rem  = F - main * 64;           // leftover 16-col strips (F % 16 == 0)
    if (main > 0) {
      k_gemm_wmma_f32<4><<<dim3(mt, main), 32, 0, stream>>>(Xin, W, hbuf, K, F, 0);
    }
    if (rem > 0) {
      k_gemm_wmma_f32<1><<<dim3(mt, rem / 16), 32, 0, stream>>>(Xin, W, hbuf, K, F,
                                                                main * 64);
    }
    int cpe = F >> 2;
    int total = N * cpe;
    k_init_out<<<(total + T - 1) / T, T, 0, stream>>>(hbuf, b, dis, Aout, N, F);
    int epb = (cpe >= 256) ? 1 : (256 / cpe);  // 4 (F=256), 2 (F=512), 1 (F=2000)
    k_edge_scatter<<<(E + epb - 1) / epb, T, 0, stream>>>(hbuf, Aout, src, dst, nrm,
                                                          E, F, epb);
  };

  layer(x,    W1, b1, abuf, D_IN, D_H1);   // A1 in abuf
  layer(abuf, W2, b2, abuf, D_H1, D_H2);   // A2 overwrites abuf (stream-ordered)
  layer(abuf, Wr, br, out,  D_H2, D_OUT);  // final layer straight into d_out
}